// PrechooseSMI_12884901888001
// MI455X (gfx1250) — compile-verified
//
#include <hip/hip_runtime.h>
#include <hip/hip_bf16.h>
#include <stdint.h>

// Problem constants (fixed by the reference)
#define BATCH 32
#define CHAN  25
#define HH    256
#define WW    256
#define PK    128     // pool window
#define OUTN  129     // 256 - 128 + 1
#define NSEL  12      // round(25/2)
#define NCH   13      // ceil(25/2)

// ---------------------------------------------------------------------------
// Kernel 1: per-(b,c) image, min over all 128x128 window sums.
// 256 threads: thread j owns column j. Running 128-row column sum in a
// register; per row-window an inclusive prefix scan over the 256 column sums
// in LDS yields every horizontal 128-wide window sum in 8 steps.
// ---------------------------------------------------------------------------
__global__ __launch_bounds__(256) void pool_min_kernel(
    const float* __restrict__ x, float* __restrict__ m)
{
    const int img = blockIdx.x;                    // b*CHAN + c
    const float* __restrict__ base = x + (size_t)img * (HH * WW);
    const int j = threadIdx.x;

    __shared__ float buf[2][256];
    __shared__ float red[256];

    // initial column window: rows 0..127
    float s = 0.f;
    #pragma unroll 8
    for (int r = 0; r < PK; ++r) s += base[r * WW + j];

    float mymin = 3.402823e38f;

    for (int i = 0; i < OUTN; ++i) {
        buf[0][j] = s;
        __syncthreads();

        // Hillis-Steele inclusive scan over 256 entries (8 steps, 2 buffers)
        int src = 0;
        #pragma unroll
        for (int off = 1; off < 256; off <<= 1) {
            float v = buf[src][j];
            if (j >= off) v += buf[src][j - off];
            buf[src ^ 1][j] = v;
            src ^= 1;
            __syncthreads();
        }
        // src is back to 0 here (8 flips); P = inclusive prefix
        if (j < OUTN) {
            float w = buf[src][j + PK - 1] - ((j > 0) ? buf[src][j - 1] : 0.f);
            mymin = fminf(mymin, w);
        }
        __syncthreads();   // protect buf[0] before next iteration's write

        if (i < OUTN - 1) {
            // slide the column window down one row
            if (i < HH - PK - 4)   // prefetch a future "add" row (L2 hint)
                __builtin_prefetch(base + (size_t)(i + PK + 4) * WW + j, 0, 1);
            s += base[(i + PK) * WW + j] - base[i * WW + j];
        }
    }

    // min-reduce across 256 threads
    red[j] = mymin;
    __syncthreads();
    #pragma unroll
    for (int off = 128; off > 0; off >>= 1) {
        if (j < off) red[j] = fminf(red[j], red[j + off]);
        __syncthreads();
    }
    if (j == 0) m[img] = red[0] * (1.0f / (float)(PK * PK));
}

// ---------------------------------------------------------------------------
// Kernel 2: per-batch selection. Tiny serial work (25 elements).
// sel_vals = ascending-sorted m[13:]; idx[k] = first channel with m==sel_vals[k];
// periods = sorted(idx).
// ---------------------------------------------------------------------------
__global__ void select_kernel(const float* __restrict__ m,
                              int* __restrict__ idx_out,
                              float* __restrict__ periods_out)
{
    const int b = blockIdx.x;
    if (threadIdx.x != 0) return;

    float v[CHAN], sv[CHAN];
    for (int c = 0; c < CHAN; ++c) { v[c] = m[b * CHAN + c]; sv[c] = v[c]; }

    // insertion sort ascending
    for (int i = 1; i < CHAN; ++i) {
        float key = sv[i];
        int k = i - 1;
        while (k >= 0 && sv[k] > key) { sv[k + 1] = sv[k]; --k; }
        sv[k + 1] = key;
    }

    int idx[NSEL];
    for (int k = 0; k < NSEL; ++k) {
        const float t = sv[NCH + k];
        int f = 0;
        for (int c = 0; c < CHAN; ++c) {
            if (v[c] == t) { f = c; break; }
        }
        idx[k] = f;
        idx_out[b * NSEL + k] = f;
    }

    // periods = sort(idx) ascending, emitted as float (output dtype)
    int p[NSEL];
    for (int k = 0; k < NSEL; ++k) p[k] = idx[k];
    for (int i = 1; i < NSEL; ++i) {
        int key = p[i];
        int k = i - 1;
        while (k >= 0 && p[k] > key) { p[k + 1] = p[k]; --k; }
        p[k + 1] = key;
    }
    for (int k = 0; k < NSEL; ++k)
        periods_out[b * NSEL + k] = (float)p[k];
}

// ---------------------------------------------------------------------------
// Kernel 3: gather-copy the 12 selected 256KB channels per batch through LDS
// using CDNA5 async global<->LDS DMA (ASYNCcnt-tracked), 4 x b128 in flight
// per thread per phase.
// ---------------------------------------------------------------------------
__global__ __launch_bounds__(256) void gather_copy_kernel(
    const float* __restrict__ x, const int* __restrict__ idx,
    float* __restrict__ out)
{
    const int bk = blockIdx.x;               // b*NSEL + k
    const int b  = bk / NSEL;
    const int ch = idx[bk];

    const float4* __restrict__ s4 =
        (const float4*)(x + ((size_t)b * CHAN + ch) * (HH * WW));
    float4* __restrict__ d4 = (float4*)(out + (size_t)bk * (HH * WW));

    __shared__ float4 stage[4][256];         // 16 KB staging
    const int t = threadIdx.x;

    // 65536 floats = 16384 float4; 256 threads * 4 chunks = 1024 float4/iter
    for (int it = 0; it < 16; ++it) {
        const int base = it * 1024 + t;
        #pragma unroll
        for (int p = 0; p < 4; ++p) {
            unsigned loff = (unsigned)(uintptr_t)(&stage[p][t]);
            const float4* gp = s4 + base + p * 256;
            asm volatile("global_load_async_to_lds_b128 %0, %1, off"
                         :: "v"(loff), "v"(gp) : "memory");
        }
        asm volatile("s_wait_asynccnt 0" ::: "memory");
        #pragma unroll
        for (int p = 0; p < 4; ++p) {
            unsigned loff = (unsigned)(uintptr_t)(&stage[p][t]);
            float4* gp = d4 + base + p * 256;
            asm volatile("global_store_async_from_lds_b128 %0, %1, off"
                         :: "v"(gp), "v"(loff) : "memory");
        }
        asm volatile("s_wait_asynccnt 0" ::: "memory");  // LDS reuse next iter
    }
}

// ---------------------------------------------------------------------------
extern "C" void kernel_launch(void* const* d_in, const int* in_sizes, int n_in,
                              void* d_out, int out_size, void* d_ws, size_t ws_size,
                              hipStream_t stream)
{
    (void)in_sizes; (void)n_in; (void)out_size; (void)ws_size;

    const float* x = (const float*)d_in[0];
    float* out = (float*)d_out;

    // workspace layout: m[800] floats at 0, idx[384] ints at +4096
    float* m   = (float*)d_ws;
    int*   idx = (int*)((char*)d_ws + 4096);

    float* periods_out = out + (size_t)BATCH * NSEL * HH * WW;

    pool_min_kernel<<<BATCH * CHAN, 256, 0, stream>>>(x, m);
    select_kernel<<<BATCH, 32, 0, stream>>>(m, idx, periods_out);
    gather_copy_kernel<<<BATCH * NSEL, 256, 0, stream>>>(x, idx, out);
}